// UpSample_11390253269667
// MI455X (gfx1250) — compile-verified
//
#include <hip/hip_runtime.h>
#include <hip/hip_bf16.h>

typedef float v2f __attribute__((ext_vector_type(2)));
typedef float v8f __attribute__((ext_vector_type(8)));

// ---------------------------------------------------------------------------
// Generic tiled GEMM / gather-sparse-conv using V_WMMA_F32_16X16X4_F32.
//   out[m, 0:64] = sum_t  x[src(t,m), 0:ci] @ w[t, 0:ci, 0:64]
// src(t,m) = nbr[t*n + m] (sentinel == n -> zero row), or m when nbr==null.
// Block: 256 threads = 8 waves; each wave computes a 16x64 tile (4 x v8f acc).
// gridDim.y selects a weight/output slice (used for the 8 up-sample taps).
// ---------------------------------------------------------------------------
__global__ __launch_bounds__(256) void spconv_wmma(
    const float* __restrict__ x, int ci,
    const float* __restrict__ w, long long w_slice,
    const int* __restrict__ nbr, int n, int taps,
    float* __restrict__ out, long long out_slice)
{
    __shared__ float lds[8192];              // ci*64 floats, paired-row layout
    w   += (long long)blockIdx.y * w_slice;
    out += (long long)blockIdx.y * out_slice;

    const int tid    = threadIdx.x;
    const int wave   = tid >> 5;
    const int lane   = tid & 31;
    const int lane_m = lane & 15;
    const int hi     = lane >> 4;            // 0: K rows {0,1}, 1: K rows {2,3}
    const int row_base = blockIdx.x * 128 + wave * 16;
    const int row      = row_base + lane_m;
    const int kcount   = ci >> 2;            // K-steps of 4
    const int welems   = ci * 64;
    const int wpairs   = welems >> 1;        // b64 chunks per tap

    v8f acc0 = {}, acc1 = {}, acc2 = {}, acc3 = {};
    const v2f vzero = {0.f, 0.f};

    for (int t = 0; t < taps; ++t) {
        __syncthreads();
        // Stage w[t] (ci x 64) into LDS pair layout with wide global loads:
        // memory pair (r,c),(r,c+1) -> LDS floats f, f+2 where
        // f = (r>>1)*128 + c*2 + (r&1). One global b64 per 2 elements.
        const float* wt = w + (long long)t * welems;
        for (int e = tid; e < wpairs; e += 256) {
            int r = e >> 5;               // row 0..ci-1
            int c = (e & 31) << 1;        // even column
            v2f val = *(const v2f*)(wt + r * 64 + c);
            int f = ((r >> 1) << 7) + (c << 1) + (r & 1);
            lds[f]     = val.x;
            lds[f + 2] = val.y;
        }
        __syncthreads();

        // Prefetch next tap's weight slab (32KB = 256 x 128B lines) into L2/L0;
        // lowers to global_prefetch_b8, no LOADcnt cost, hides staging latency.
        if (t + 1 < taps)
            __builtin_prefetch(wt + welems + tid * 32, 0, 1);

        int src = n;
        if (row < n) src = nbr ? nbr[(long long)t * n + row] : row;
        const bool valid = (src >= 0) && (src < n);
        if (__ballot(valid ? 1 : 0) == 0) continue;    // wave-uniform skip

        const int src_c = valid ? src : 0;             // keep loads in-bounds
        const float* rowp = x + (long long)src_c * ci + (hi << 1);

        for (int kk = 0; kk < kcount; ++kk) {
            v2f a = *(const v2f*)(rowp + (kk << 2));
            a = valid ? a : vzero;                     // cndmask, EXEC stays full
            const int pbase = (((kk << 1) + hi) << 7); // pair (2kk+hi) * 128
            v2f b0 = *(const v2f*)(lds + pbase + ((     lane_m) << 1));
            v2f b1 = *(const v2f*)(lds + pbase + ((16 + lane_m) << 1));
            v2f b2 = *(const v2f*)(lds + pbase + ((32 + lane_m) << 1));
            v2f b3 = *(const v2f*)(lds + pbase + ((48 + lane_m) << 1));
            acc0 = __builtin_amdgcn_wmma_f32_16x16x4_f32(false, a, false, b0, (short)0, acc0, false, false);
            acc1 = __builtin_amdgcn_wmma_f32_16x16x4_f32(false, a, false, b1, (short)0, acc1, false, false);
            acc2 = __builtin_amdgcn_wmma_f32_16x16x4_f32(false, a, false, b2, (short)0, acc2, false, false);
            acc3 = __builtin_amdgcn_wmma_f32_16x16x4_f32(false, a, false, b3, (short)0, acc3, false, false);
        }
    }

    // C/D layout: VGPR r holds M = r + 8*hi, N = lane_m (per 16-col tile)
    for (int r = 0; r < 8; ++r) {
        int m = row_base + (hi << 3) + r;
        if (m < n) {
            long long o = (long long)m * 64 + lane_m;
            out[o]      = acc0[r];
            out[o + 16] = acc1[r];
            out[o + 32] = acc2[r];
            out[o + 48] = acc3[r];
        }
    }
}

// ---------------------------------------------------------------------------
// x[n, 0:64]  = y_up[up_kidx[n], up_src[n], :]   (raw, pre-BN)
// x[n, 64:128]= feats2[n, :]
// ---------------------------------------------------------------------------
__global__ void gather_concat(const float* __restrict__ yup,
                              const float* __restrict__ feats2,
                              const int* __restrict__ up_src,
                              const int* __restrict__ up_kidx,
                              float* __restrict__ xcat, int n1, int n2)
{
    int idx = blockIdx.x * 256 + threadIdx.x;
    if (idx >= n2 * 128) return;
    int n = idx >> 7, c = idx & 127;
    float v;
    if (c < 64) {
        int s = up_src[n];
        int k = up_kidx[n];
        v = yup[((long long)k * n1 + s) * 64 + c];
    } else {
        v = feats2[(long long)n * 64 + (c - 64)];
    }
    xcat[idx] = v;
}

// ---------------------------------------------------------------------------
// Deterministic per-channel sum / sum-of-squares over n rows (64 channels).
// One block per channel; fixed-order tree reduction.
// ---------------------------------------------------------------------------
__global__ void bn_stats(const float* __restrict__ src, int n, int stride,
                         float* __restrict__ sums)
{
    const int c = blockIdx.x;     // 0..63
    float s = 0.f, q = 0.f;
    for (int i = threadIdx.x; i < n; i += 256) {
        float v = src[(long long)i * stride + c];
        s += v;
        q += v * v;
    }
    __shared__ float sh[512];
    sh[threadIdx.x]       = s;
    sh[256 + threadIdx.x] = q;
    __syncthreads();
    for (int off = 128; off > 0; off >>= 1) {
        if (threadIdx.x < off) {
            sh[threadIdx.x]       += sh[threadIdx.x + off];
            sh[256 + threadIdx.x] += sh[256 + threadIdx.x + off];
        }
        __syncthreads();
    }
    if (threadIdx.x == 0) {
        sums[c]      = sh[0];
        sums[64 + c] = sh[256];
    }
}

// ---------------------------------------------------------------------------
// In-place BN (training-mode, biased var) + ReLU on the first 64 channels of
// a row-major [n, stride] buffer.
// ---------------------------------------------------------------------------
__global__ void bn_apply(float* __restrict__ buf, int n, int stride,
                         const float* __restrict__ gamma,
                         const float* __restrict__ beta,
                         const float* __restrict__ sums)
{
    int idx = blockIdx.x * 256 + threadIdx.x;
    if (idx >= n * 64) return;
    int c = idx & 63;
    long long r = idx >> 6;
    float inv_n = 1.f / (float)n;
    float mean  = sums[c] * inv_n;
    float var   = sums[64 + c] * inv_n - mean * mean;
    float sc    = gamma[c] * rsqrtf(var + 1e-5f);
    float shv   = beta[c] - mean * sc;
    float* p = buf + r * stride + c;
    float v = *p * sc + shv;
    *p = v > 0.f ? v : 0.f;
}

// ---------------------------------------------------------------------------
extern "C" void kernel_launch(void* const* d_in, const int* in_sizes, int n_in,
                              void* d_out, int out_size, void* d_ws, size_t ws_size,
                              hipStream_t stream)
{
    const int CIN  = 128;
    const int N1   = in_sizes[0] / CIN;        // 15000
    const int N2   = in_sizes[1] / (CIN / 2);  // 60000

    const float* feats1   = (const float*)d_in[0];
    const float* feats2   = (const float*)d_in[1];
    const float* w_up     = (const float*)d_in[2];   // [8,128,64]
    const float* gamma_up = (const float*)d_in[3];
    const float* beta_up  = (const float*)d_in[4];
    const float* w1       = (const float*)d_in[5];   // [27,128,64]
    const float* gamma1   = (const float*)d_in[6];
    const float* beta1    = (const float*)d_in[7];
    const float* w2       = (const float*)d_in[8];   // [27,64,64]
    const float* gamma2   = (const float*)d_in[9];
    const float* beta2    = (const float*)d_in[10];
    const int*   up_src   = (const int*)d_in[11];
    const int*   up_kidx  = (const int*)d_in[12];
    const int*   nbr_src  = (const int*)d_in[13];    // [27, N2], sentinel = N2
    // d_in[14] (nbr_dst) not needed: scatter == identity gather (see analysis)

    float* out = (float*)d_out;

    // Workspace layout (floats): x[N2*128] | yup[8*N1*64] (reused as h[N2*64]) | stats[128]
    float* ws_x     = (float*)d_ws;
    float* ws_yup   = ws_x + (size_t)N2 * 128;
    float* ws_h     = ws_yup;                         // reuse after gather_concat
    float* ws_stats = ws_yup + (size_t)8 * N1 * 64;

    dim3 blk(256);

    // 1) up-sample GEMM: yup[k] = feats1 @ w_up[k]   (k via gridDim.y)
    spconv_wmma<<<dim3((N1 + 127) / 128, 8), blk, 0, stream>>>(
        feats1, 128, w_up, (long long)128 * 64, nullptr, N1, 1,
        ws_yup, (long long)N1 * 64);

    // 2) gather parent features + concat feats2 -> x (x[:, :64] still raw)
    gather_concat<<<(N2 * 128 + 255) / 256, blk, 0, stream>>>(
        ws_yup, feats2, up_src, up_kidx, ws_x, N1, N2);

    // 3) BN+ReLU on x[:, :64]
    bn_stats<<<64, blk, 0, stream>>>(ws_x, N2, 128, ws_stats);
    bn_apply<<<(N2 * 64 + 255) / 256, blk, 0, stream>>>(
        ws_x, N2, 128, gamma_up, beta_up, ws_stats);

    // 4) sparse conv1: h = sum_t x[nbr_src[t]] @ w1[t]
    spconv_wmma<<<dim3((N2 + 127) / 128, 1), blk, 0, stream>>>(
        ws_x, 128, w1, 0, nbr_src, N2, 27, ws_h, 0);
    bn_stats<<<64, blk, 0, stream>>>(ws_h, N2, 64, ws_stats);
    bn_apply<<<(N2 * 64 + 255) / 256, blk, 0, stream>>>(
        ws_h, N2, 64, gamma1, beta1, ws_stats);

    // 5) sparse conv2 -> d_out (raw), then BN+ReLU in place
    spconv_wmma<<<dim3((N2 + 127) / 128, 1), blk, 0, stream>>>(
        ws_h, 64, w2, 0, nbr_src, N2, 27, out, 0);
    bn_stats<<<64, blk, 0, stream>>>(out, N2, 64, ws_stats);
    bn_apply<<<(N2 * 64 + 255) / 256, blk, 0, stream>>>(
        out, N2, 64, gamma2, beta2, ws_stats);
}